// BTGRule_21921513079260
// MI455X (gfx1250) — compile-verified
//
#include <hip/hip_runtime.h>
#include <hip/hip_bf16.h>
#include <math.h>

// Problem constants from the reference.
#define N1   65                 // (N+1)
#define HID  512
#define M_ROWS (N1 * N1)        // 4225 flattened chart rows
#define M_TILES ((M_ROWS + 15) / 16)   // 265
#define M_PAD  (M_TILES * 16)          // 4240 (padded rows in workspace)
#define N_TILES (HID / 16)             // 32

typedef __attribute__((ext_vector_type(2))) float v2f;
typedef __attribute__((ext_vector_type(8))) float v8f;

// ---------------------------------------------------------------------------
// Stage 1: L = X @ Wl + bl ; R = X @ Wr + br  via V_WMMA_F32_16X16X4_F32.
// One wave (32 threads) per 16x16 output tile; grid.z selects Wl/Wr.
// VGPR layouts per CDNA5 ISA 7.12.2:
//   A 16x4 f32 : lane(lm,half) holds A[lm, k0+2*half], A[lm, k0+2*half+1]
//   B 4x16 f32 : lane(lm,half) holds B[k0+2*half, lm], B[k0+2*half+1, lm]
//   D 16x16    : vgpr r at lane(lm,half) = D[r + 8*half, lm]
// Output buffers are padded to M_PAD rows so the epilogue stores are
// branch-free (no per-row EXEC manipulation); pad rows are never read.
// ---------------------------------------------------------------------------
__global__ void btg_proj_wmma(const float* __restrict__ X,
                              const float* __restrict__ Wl,
                              const float* __restrict__ bl,
                              const float* __restrict__ Wr,
                              const float* __restrict__ br,
                              float* __restrict__ Lout,
                              float* __restrict__ Rout) {
    const int tm = blockIdx.x;          // M tile: 0..264
    const int tn = blockIdx.y;          // N tile: 0..31
    const float* __restrict__ W    = (blockIdx.z == 0) ? Wl : Wr;
    const float* __restrict__ bias = (blockIdx.z == 0) ? bl : br;
    float* __restrict__ Out        = (blockIdx.z == 0) ? Lout : Rout;

    const int lane = threadIdx.x;       // 0..31, EXEC all-ones (no divergence)
    const int half = lane >> 4;         // 0 or 1
    const int lm   = lane & 15;

    const int row  = tm * 16 + lm;
    const int rowc = (row < M_ROWS) ? row : (M_ROWS - 1); // branch-free clamp for loads
    const float* __restrict__ Arow = X + (size_t)rowc * HID;
    const int colbase = tn * 16;

    v8f acc = {};
#pragma unroll 4
    for (int k0 = 0; k0 < HID; k0 += 4) {
        const int ka = k0 + 2 * half;
        v2f a = *(const v2f*)(Arow + ka);               // A[lm, ka], A[lm, ka+1]
        v2f b;
        b.x = W[(size_t)ka * HID + colbase + lm];       // B[ka,   col]
        b.y = W[(size_t)(ka + 1) * HID + colbase + lm]; // B[ka+1, col]
        // 8 args: (neg_a, A, neg_b, B, c_mod, C, reuse_a, reuse_b)
        acc = __builtin_amdgcn_wmma_f32_16x16x4_f32(
            false, a, false, b, (short)0, acc, false, false);
    }

    const float bv = bias[colbase + lm];
    float* __restrict__ op =
        Out + (size_t)(tm * 16 + 8 * half) * HID + colbase + lm;
#pragma unroll
    for (int r = 0; r < 8; ++r) {
        op[(size_t)r * HID] = acc[r] + bv;              // unconditional, padded ws
    }
}

// ---------------------------------------------------------------------------
// Stage 2a: zero the dense [65,65,65,2] output (invalid triples stay 0).
// ---------------------------------------------------------------------------
__global__ void btg_zero(float* __restrict__ out, int n) {
    int idx = blockIdx.x * blockDim.x + threadIdx.x;
    if (idx < n) out[idx] = 0.0f;
}

// ---------------------------------------------------------------------------
// Stage 2b: fused tanh + (512 -> 2) contraction over VALID triples only.
// One block per (i,j) with i<j<64; L[i,j,:] and Wout staged in LDS.
// One wave per split point k (strided); lanes parallel over h; tanhf lowers
// to the v_tanh_f32 transcendental; wave32 __shfl_xor reduction.
// ---------------------------------------------------------------------------
__global__ void btg_score(const float* __restrict__ Lp,
                          const float* __restrict__ Rp,
                          const float* __restrict__ Wout,
                          const float* __restrict__ bout,
                          float* __restrict__ out) {
    const int bid = blockIdx.x;
    const int i = bid / N1;
    const int j = bid % N1;
    if (!(i < j && j < N1 - 1)) return;   // uniform per block: safe before barrier

    __shared__ float Ls[HID];
    __shared__ float Wo[HID * 2];

    const float* __restrict__ Lrow = Lp + (size_t)(i * N1 + j) * HID;
    for (int h = threadIdx.x; h < HID; h += blockDim.x)     Ls[h] = Lrow[h];
    for (int h = threadIdx.x; h < HID * 2; h += blockDim.x) Wo[h] = Wout[h];
    __syncthreads();

    const int wave = threadIdx.x >> 5;
    const int lane = threadIdx.x & 31;
    const int nwaves = blockDim.x >> 5;
    const float b0 = bout[0], b1 = bout[1];

    for (int k = j + 1 + wave; k < N1; k += nwaves) {
        const float* __restrict__ Rrow = Rp + (size_t)(j * N1 + k) * HID;
        float a0 = 0.0f, a1 = 0.0f;
#pragma unroll 4
        for (int h = lane; h < HID; h += 32) {
            float t = tanhf(Ls[h] + Rrow[h]);          // v_tanh_f32
            a0 = fmaf(t, Wo[2 * h],     a0);
            a1 = fmaf(t, Wo[2 * h + 1], a1);
        }
#pragma unroll
        for (int off = 16; off > 0; off >>= 1) {       // wave32 reduce
            a0 += __shfl_xor(a0, off, 32);
            a1 += __shfl_xor(a1, off, 32);
        }
        if (lane == 0) {
            size_t o = ((size_t)(i * N1 + j) * N1 + k) * 2;
            out[o]     = a0 + b0;
            out[o + 1] = a1 + b1;
        }
    }
}

// ---------------------------------------------------------------------------
extern "C" void kernel_launch(void* const* d_in, const int* in_sizes, int n_in,
                              void* d_out, int out_size, void* d_ws, size_t ws_size,
                              hipStream_t stream) {
    const float* span = (const float*)d_in[0];  // [65,65,512]
    const float* Wl   = (const float*)d_in[1];  // [512,512]
    const float* bl   = (const float*)d_in[2];  // [512]
    const float* Wr   = (const float*)d_in[3];  // [512,512]
    const float* br   = (const float*)d_in[4];  // [512]
    const float* Wout = (const float*)d_in[5];  // [512,2]
    const float* bout = (const float*)d_in[6];  // [2]
    float* out = (float*)d_out;                 // [65,65,65,2]

    // Workspace: L then R, each padded to M_PAD rows (~17.4 MB total).
    float* Lp = (float*)d_ws;
    float* Rp = Lp + (size_t)M_PAD * HID;

    // Stage 1: both projections via WMMA f32.
    dim3 g1(M_TILES, N_TILES, 2);
    btg_proj_wmma<<<g1, 32, 0, stream>>>(span, Wl, bl, Wr, br, Lp, Rp);

    // Stage 2a: zero dense output (invalid i,j,k stay exactly 0).
    const int nOut = N1 * N1 * N1 * 2;  // 549,250
    btg_zero<<<(nOut + 255) / 256, 256, 0, stream>>>(out, nOut);

    // Stage 2b: fused tanh + score over valid triples only.
    btg_score<<<N1 * N1, 256, 0, stream>>>(Lp, Rp, Wout, bout, out);
}